// LeftPooling_70927089926349
// MI455X (gfx1250) — compile-verified
//
#include <hip/hip_runtime.h>

// Reverse (suffix) cumulative max along axis 1 of a [B=16, H=128, W=128, C=256]
// fp32 tensor. Pure streaming problem: 256 MiB read + 256 MiB write, ~23 us
// floor at 23.3 TB/s HBM. One thread owns one float4 column (b, w, c4) and
// scans h=127..0 keeping a running componentwise max. WMMA is inapplicable
// (max is non-linear), so the optimized path is the data-movement path:
// wave32, b128 non-temporal loads/stores, global_prefetch_b8 ahead of the
// strided walk.

typedef float f32x4 __attribute__((ext_vector_type(4)));

__device__ __forceinline__ f32x4 vmax4(f32x4 a, f32x4 b) {
    f32x4 r;
    r.x = fmaxf(a.x, b.x);
    r.y = fmaxf(a.y, b.y);
    r.z = fmaxf(a.z, b.z);
    r.w = fmaxf(a.w, b.w);
    return r;
}

namespace {
constexpr unsigned B = 16;
constexpr unsigned H = 128;                 // scanned axis
constexpr unsigned W = 128;
constexpr unsigned C = 256;
constexpr unsigned STRIDE = W * C / 4;      // float4 stride between h-slices = 8192 (2^13)
constexpr unsigned NCOLS  = B * STRIDE;     // independent float4 scan columns = 131072
constexpr int      UNROLL = 8;              // b128 loads in flight per thread per body
constexpr int      PF_DIST = 8;             // prefetch distance in h-slices (~16 MiB aggregate window)
constexpr unsigned BLOCK = 256;             // 8 wave32s per workgroup
}

__global__ __launch_bounds__(BLOCK) void revcummax_kernel(
    const f32x4* __restrict__ in, f32x4* __restrict__ out)
{
    // Grid exactly covers NCOLS (exact multiple of BLOCK): no bounds check.
    const unsigned col = blockIdx.x * BLOCK + threadIdx.x;

    // STRIDE = 2^13 and col is unsigned: shift + mask, no sign fixup.
    const unsigned b  = col >> 13;
    const unsigned cw = col & (STRIDE - 1u);
    const size_t base = (size_t)b * H * STRIDE + cw;

    const f32x4* pin  = in  + base;
    f32x4*       pout = out + base;

    const float ninf = -__builtin_inff();
    f32x4 m = {ninf, ninf, ninf, ninf};

    // One lane per 8 issues the prefetch: a wave covers 512 contiguous bytes,
    // so 4 lanes still touch every cacheline without 8x duplicate requests.
    const bool pf_lane = (threadIdx.x & 7u) == 0u;

    for (int h0 = (int)H - UNROLL; h0 >= 0; h0 -= UNROLL) {
        const int hp = h0 - PF_DIST;
        if (pf_lane && hp >= 0) {
            __builtin_prefetch((const void*)(pin + (size_t)hp * STRIDE), 0, 1);
        }

        // Issue all UNROLL b128 loads before the serial max chain so the
        // memory pipe stays full despite the scan dependency.
        f32x4 v[UNROLL];
#pragma unroll
        for (int i = 0; i < UNROLL; ++i) {
            v[i] = __builtin_nontemporal_load(pin + (size_t)(h0 + i) * STRIDE);
        }

        // Running suffix max: consume in decreasing-h order.
#pragma unroll
        for (int i = UNROLL - 1; i >= 0; --i) {
            m = vmax4(m, v[i]);
            __builtin_nontemporal_store(m, pout + (size_t)(h0 + i) * STRIDE);
        }
    }
}

extern "C" void kernel_launch(void* const* d_in, const int* in_sizes, int n_in,
                              void* d_out, int out_size, void* d_ws, size_t ws_size,
                              hipStream_t stream) {
    (void)in_sizes; (void)n_in; (void)out_size; (void)d_ws; (void)ws_size;
    const f32x4* in  = (const f32x4*)d_in[0];
    f32x4*       out = (f32x4*)d_out;

    dim3 grid(NCOLS / BLOCK);   // 512 blocks
    dim3 block(BLOCK);          // 256 threads = 8 wave32s
    hipLaunchKernelGGL(revcummax_kernel, grid, block, 0, stream, in, out);
}